// FourierSeasonalExtractor_43181601194288
// MI455X (gfx1250) — compile-verified
//
#include <hip/hip_runtime.h>

typedef __attribute__((ext_vector_type(2))) float v2f;
typedef __attribute__((ext_vector_type(4))) float v4f;
typedef __attribute__((ext_vector_type(8))) float v8f;

#define NROWS 4096
#define SLEN  8192
// rfft bins kept by the harmonic mask for S=8192, period=24, harmonics 1..3
#define KBIN0 341
#define KBIN1 683
#define KBIN2 1024
#define TWO_PI_OVER_S 7.6699039210050492e-4f  // 2*pi/8192

// basis_j(t): j=0..2 -> cos(2*pi*k_j*t/S), j=3..5 -> sin(...), j>=6 -> 0 (pad to N=16)
__device__ __forceinline__ float basis_val(int t, int n) {
    if (n >= 6) return 0.0f;
    const int j = (n >= 3) ? (n - 3) : n;
    const int kbin = (j == 0) ? KBIN0 : ((j == 1) ? KBIN1 : KBIN2);
    const int m = (kbin * t) & (SLEN - 1);          // exact modular phase
    const float ang = (float)m * TWO_PI_OVER_S;     // in [0, 2*pi)
    return (n < 3) ? __cosf(ang) : __sinf(ang);
}

// ---------------------------------------------------------------------------
// Kernel 0: build pairs-table TB[p][n] = {basis_n(2p), basis_n(2p+1)},
// laid out so a WMMA B-operand lane (v2f) is one b64 load.
// 4096 pairs x 16 cols x 8B = 512 KB (L2 resident).
// ---------------------------------------------------------------------------
__global__ void __launch_bounds__(256)
k_basis(v2f* __restrict__ tb) {
    const int idx = blockIdx.x * 256 + threadIdx.x;   // 65536 entries
    const int p = idx >> 4;
    const int n = idx & 15;
    v2f v;
    v.x = basis_val(2 * p,     n);
    v.y = basis_val(2 * p + 1, n);
    tb[idx] = v;
}

// ---------------------------------------------------------------------------
// Kernel 1: coefficients C[16 rows x 16 cols] per block via fp32 WMMA.
// A (16x4 f32): lane l holds rows M=l&15, K = 2*(l>>4) + {0,1}  -> one b64 load
// B (4x16 f32): lane l holds N=l&15,   K = 2*(l>>4) + {0,1}     -> one b64 table load
// 4 waves split K=8192; LDS reduce; only N=0..5 are meaningful.
// ---------------------------------------------------------------------------
__global__ void __launch_bounds__(128)
k_coef(const float* __restrict__ X, const v2f* __restrict__ tb,
       float* __restrict__ coef) {
    __shared__ float red[4][32][8];

    const int lane = threadIdx.x & 31;
    const int wave = threadIdx.x >> 5;
    const int n    = lane & 15;
    const int hi   = lane >> 4;
    const int rowBlock = blockIdx.x * 16;
    const int kStart   = wave * (SLEN / 4);           // 2048 per wave

    // A pointer: row = n (M index), cols kStart + 2*hi + {0,1} + k
    const v2f* ap = (const v2f*)(X + (size_t)(rowBlock + n) * SLEN + kStart + 2 * hi);
    // B pointer: table entry ((kStart+k)/2 + hi)*16 + n
    const v2f* bp = tb + ((size_t)(kStart >> 1) + hi) * 16 + n;

    v8f c = {};
#pragma unroll 4
    for (int k = 0; k < SLEN / 4; k += 4) {
        v2f a = ap[k >> 1];
        v2f b = bp[(k >> 1) * 16];
        c = __builtin_amdgcn_wmma_f32_16x16x4_f32(
                /*neg_a=*/false, a, /*neg_b=*/false, b,
                /*c_mod=*/(short)0, c, /*reuse_a=*/false, /*reuse_b=*/false);
    }

#pragma unroll
    for (int r = 0; r < 8; ++r) red[wave][lane][r] = c[r];
    __syncthreads();

    if (threadIdx.x < 32) {
#pragma unroll
        for (int r = 0; r < 8; ++r) {
            const float s = red[0][lane][r] + red[1][lane][r] +
                            red[2][lane][r] + red[3][lane][r];
            const int m = r + 8 * hi;                 // C/D layout: M = r + 8*(lane>=16)
            coef[(size_t)(rowBlock + m) * 16 + n] = s;
        }
    }
}

// ---------------------------------------------------------------------------
// Kernel 2: streaming synthesis. 1 float4 of one row per lane.
// seasonal = (2/S) * sum_j a_j * basis_j(t);  residual = x - seasonal.
// NT stores for the 256 MB of output keep X resident in L2 from pass 1.
// ---------------------------------------------------------------------------
__global__ void __launch_bounds__(256)
k_synth(const float* __restrict__ X, const float* __restrict__ coef,
        float* __restrict__ seas, float* __restrict__ resid) {
    const int gid = blockIdx.x * 256 + threadIdx.x;
    const int row = gid >> 11;                        // 2048 float4 per row (block-uniform)
    const int t0  = (gid & 2047) << 2;

    const float sc = 2.0f / (float)SLEN;
    const float* cr = coef + (size_t)row * 16;
    const float a0 = cr[0] * sc, a1 = cr[1] * sc, a2 = cr[2] * sc;
    const float a3 = cr[3] * sc, a4 = cr[4] * sc, a5 = cr[5] * sc;

    const v4f xv = ((const v4f*)X)[gid];
    v4f se, re;
#pragma unroll
    for (int i = 0; i < 4; ++i) {
        const int t  = t0 + i;
        const int m0 = (KBIN0 * t) & (SLEN - 1);
        const int m1 = (KBIN1 * t) & (SLEN - 1);
        const int m2 = (KBIN2 * t) & (SLEN - 1);
        float s0, c0, s1, c1, s2, c2;
        __sincosf((float)m0 * TWO_PI_OVER_S, &s0, &c0);
        __sincosf((float)m1 * TWO_PI_OVER_S, &s1, &c1);
        __sincosf((float)m2 * TWO_PI_OVER_S, &s2, &c2);
        const float sv = a0 * c0 + a1 * c1 + a2 * c2 +
                         a3 * s0 + a4 * s1 + a5 * s2;
        se[i] = sv;
        re[i] = xv[i] - sv;
    }
    __builtin_nontemporal_store(se, (v4f*)seas  + gid);
    __builtin_nontemporal_store(re, (v4f*)resid + gid);
}

// ---------------------------------------------------------------------------
extern "C" void kernel_launch(void* const* d_in, const int* in_sizes, int n_in,
                              void* d_out, int out_size, void* d_ws, size_t ws_size,
                              hipStream_t stream) {
    (void)in_sizes; (void)n_in; (void)out_size; (void)ws_size;

    const float* X = (const float*)d_in[0];           // [4096, 8192] f32
    float* seasonal = (float*)d_out;                  // [4096, 8192]
    float* residual = seasonal + (size_t)NROWS * SLEN;

    // Workspace layout: [0, 512KB) basis pairs-table, [512KB, 768KB) coefficients
    v2f*   tb   = (v2f*)d_ws;                                  // 4096*16 v2f
    float* coef = (float*)((char*)d_ws + (size_t)512 * 1024);  // 4096*16 f32

    // Pass 0: basis table (65536 entries)
    k_basis<<<256, 256, 0, stream>>>(tb);
    // Pass 1: rank-6 projection coefficients via fp32 WMMA (256 WGs x 4 waves)
    k_coef<<<NROWS / 16, 128, 0, stream>>>(X, tb, coef);
    // Pass 2: streaming synthesis + residual (32768 WGs)
    k_synth<<<(int)(((size_t)NROWS * SLEN) / 1024), 256, 0, stream>>>(
        X, coef, seasonal, residual);
}